// FirstEncoder_1941325218150
// MI455X (gfx1250) — compile-verified
//
#include <hip/hip_runtime.h>
#include <cstdint>
#include <cstddef>

typedef __attribute__((ext_vector_type(2))) float v2f;
typedef __attribute__((ext_vector_type(8))) float v8f;

#define FDIM 256
#define KCHUNK 32

// ---------------------------------------------------------------- utilities
__global__ __launch_bounds__(256) void fill_kernel(float4* __restrict__ p, float v, int n4) {
  int i = blockIdx.x * blockDim.x + threadIdx.x;
  if (i < n4) p[i] = make_float4(v, v, v, v);
}

__global__ __launch_bounds__(256) void deg_kernel(const long long* __restrict__ dst,
                                                  float* __restrict__ deg, int E) {
  int i = blockIdx.x * blockDim.x + threadIdx.x;
  if (i < E) atomicAdd(&deg[(int)dst[i]], 1.0f);
}

__global__ __launch_bounds__(256) void rsqrt_kernel(float* __restrict__ d, int n) {
  int i = blockIdx.x * blockDim.x + threadIdx.x;
  if (i < n) { float x = d[i]; d[i] = (x > 0.0f) ? rsqrtf(x) : 0.0f; }
}

// ---------------------------------------------------------------- fp32 WMMA GEMM
// C[nrows x 256] = A[nrows x 256] @ W[256 x 256] (+ bias if non-null)
// Block: 256 threads = 8 waves; block tile 64 rows x 256 cols.
// Wave w: rows (w&1)*32 (+0,+16 tiles), cols (w>>1)*64 (4 tiles) -> 8 v8f accumulators.
// W staged through LDS in 32x256 K-chunks, PRE-SWIZZLED into B-fragment order so each
// B fragment is a single aligned ds_load_b64:
//   float2 {W[kc+4g+2h][c], W[kc+4g+2h+1][c]} at float index ((g*2+h)*256 + c)*2
__global__ __launch_bounds__(256) void gemm_wmma_f32(const float* __restrict__ A,
                                                     const float* __restrict__ W,
                                                     const float* __restrict__ bias,
                                                     float* __restrict__ C,
                                                     int nrows) {
  __shared__ float wlds[KCHUNK * FDIM];  // 32 KB, fragment-swizzled
  const int tid      = threadIdx.x;
  const int wave     = tid >> 5;
  const int lane     = tid & 31;
  const int laneHalf = lane >> 4;   // selects K-half for A/B frags, M-half for C
  const int laneMod  = lane & 15;
  const int rowBase  = blockIdx.x * 64 + (wave & 1) * 32;
  const int colBase  = (wave >> 1) * 64;

  v8f acc[2][4] = {};

  // A rows for this lane (clamped; clamped rows never stored)
  int ar0 = rowBase + laneMod;      if (ar0 >= nrows) ar0 = nrows - 1;
  int ar1 = rowBase + 16 + laneMod; if (ar1 >= nrows) ar1 = nrows - 1;
  const float* ap0 = A + (size_t)ar0 * FDIM + laneHalf * 2;
  const float* ap1 = A + (size_t)ar1 * FDIM + laneHalf * 2;

  for (int kc = 0; kc < FDIM; kc += KCHUNK) {
    // stage W[kc..kc+31][:] into LDS in fragment order:
    // 16 row-pairs x 128 col-pairs; each item: 2x global float2 -> 1x ds_store_b128
    for (int i = tid; i < 16 * 128; i += 256) {
      const int rp = i >> 7;          // row-pair: rows 2rp, 2rp+1  (g*2+h == rp)
      const int c  = (i & 127) * 2;   // col-pair
      const float2 w0 = *(const float2*)(W + (size_t)(kc + rp * 2 + 0) * FDIM + c);
      const float2 w1 = *(const float2*)(W + (size_t)(kc + rp * 2 + 1) * FDIM + c);
      *(float4*)(&wlds[(rp * FDIM + c) * 2]) = make_float4(w0.x, w1.x, w0.y, w1.y);
    }
    __syncthreads();

#pragma unroll
    for (int k = 0; k < KCHUNK; k += 4) {
      // A fragments (16x4 f32): lane holds A[M=laneMod][K = k + laneHalf*2 + {0,1}]
      v2f a0, a1;
      { const float2 t0 = *(const float2*)(ap0 + kc + k); a0.x = t0.x; a0.y = t0.y; }
      { const float2 t1 = *(const float2*)(ap1 + kc + k); a1.x = t1.x; a1.y = t1.y; }
      const int g2h = (k >> 1) + laneHalf;   // g*2 + h
#pragma unroll
      for (int t = 0; t < 4; ++t) {
        // B fragment: single b64 load of {W[kr][col], W[kr+1][col]}
        const int col = colBase + t * 16 + laneMod;
        const v2f b = *(const v2f*)(&wlds[(g2h * FDIM + col) * 2]);
        acc[0][t] = __builtin_amdgcn_wmma_f32_16x16x4_f32(
            false, a0, false, b, (short)0, acc[0][t], false, false);
        acc[1][t] = __builtin_amdgcn_wmma_f32_16x16x4_f32(
            false, a1, false, b, (short)0, acc[1][t], false, false);
      }
    }
    __syncthreads();
  }

  // C layout: vgpr v, lanes0-15 -> M=v, lanes16-31 -> M=v+8; N=laneMod
#pragma unroll
  for (int t = 0; t < 4; ++t) {
    const int col = colBase + t * 16 + laneMod;
    const float bv = bias ? bias[col] : 0.0f;
#pragma unroll
    for (int rt = 0; rt < 2; ++rt) {
#pragma unroll
      for (int v = 0; v < 8; ++v) {
        const int row = rowBase + rt * 16 + laneHalf * 8 + v;
        if (row < nrows) C[(size_t)row * FDIM + col] = acc[rt][t][v] + bv;
      }
    }
  }
}

// ---------------------------------------------------------------- edge scatter
// One wave per edge: agg[dst] += h2[src] * dis[src]*dis[dst]   (256 floats/row)
__global__ __launch_bounds__(256) void scatter_kernel(const long long* __restrict__ src,
                                                      const long long* __restrict__ dst,
                                                      const float* __restrict__ h2,
                                                      const float* __restrict__ dis,
                                                      float* __restrict__ agg, int E) {
  const int lane = threadIdx.x & 31;
  int w = (blockIdx.x * blockDim.x + threadIdx.x) >> 5;
  const int nw = (gridDim.x * blockDim.x) >> 5;
  for (int e = w; e < E; e += nw) {
    const int s = (int)src[e];
    const int d = (int)dst[e];
    const float norm = dis[s] * dis[d];
    const float4* hp = (const float4*)(h2 + (size_t)s * FDIM);
    float* op = agg + (size_t)d * FDIM;
#pragma unroll
    for (int it = 0; it < FDIM / 128; ++it) {
      const int j = it * 32 + lane;
      float4 v = hp[j];
      atomicAdd(op + j * 4 + 0, v.x * norm);
      atomicAdd(op + j * 4 + 1, v.y * norm);
      atomicAdd(op + j * 4 + 2, v.z * norm);
      atomicAdd(op + j * 4 + 3, v.w * norm);
    }
  }
}

// ---------------------------------------------------------------- layer epilogue
// h = h + agg + h2*dis^2 (self-loop) + bias ; optional leaky-relu on final layer
__global__ __launch_bounds__(256) void epilogue_kernel(float* __restrict__ h,
                                                       const float* __restrict__ h2,
                                                       const float* __restrict__ agg,
                                                       const float* __restrict__ dis,
                                                       const float* __restrict__ bias,
                                                       int n4, int apply_lrelu) {
  int i = blockIdx.x * blockDim.x + threadIdx.x;
  if (i >= n4) return;
  const int row = i >> 6;          // 64 float4 per 256-wide row
  const int col = (i & 63) * 4;
  const float di = dis[row];
  const float d2 = di * di;
  float4 hv = ((const float4*)h)[i];
  float4 av = ((const float4*)agg)[i];
  float4 gv = ((const float4*)h2)[i];
  float4 bv = *(const float4*)(bias + col);
  float4 r;
  r.x = hv.x + av.x + gv.x * d2 + bv.x;
  r.y = hv.y + av.y + gv.y * d2 + bv.y;
  r.z = hv.z + av.z + gv.z * d2 + bv.z;
  r.w = hv.w + av.w + gv.w * d2 + bv.w;
  if (apply_lrelu) {
    r.x = r.x > 0.0f ? r.x : 0.1f * r.x;
    r.y = r.y > 0.0f ? r.y : 0.1f * r.y;
    r.z = r.z > 0.0f ? r.z : 0.1f * r.z;
    r.w = r.w > 0.0f ? r.w : 0.1f * r.w;
  }
  ((float4*)h)[i] = r;
}

// ---------------------------------------------------------------- launch
extern "C" void kernel_launch(void* const* d_in, const int* in_sizes, int n_in,
                              void* d_out, int out_size, void* d_ws, size_t ws_size,
                              hipStream_t stream) {
  (void)n_in; (void)out_size; (void)ws_size;
  const float*     x  = (const float*)d_in[0];
  const long long* ei = (const long long*)d_in[1];
  const float*     Wp = (const float*)d_in[2];
  const float*     bp = (const float*)d_in[3];
  const float*     W1 = (const float*)d_in[4];
  const float*     b1 = (const float*)d_in[5];
  const float*     W2 = (const float*)d_in[6];
  const float*     b2 = (const float*)d_in[7];

  const int N = in_sizes[0] / FDIM;
  const int E = in_sizes[1] / 2;
  const long long* src = ei;        // edge_index[0]
  const long long* dst = ei + E;    // edge_index[1]

  float* h = (float*)d_out;         // running node features live in d_out

  char* ws = (char*)d_ws;
  float* dis = (float*)ws;                                // N floats (deg -> rsqrt in place)
  size_t off = ((size_t)N * 4 + 255) & ~(size_t)255;
  float* h2  = (float*)(ws + off);                        // N*256 floats
  float* agg = (float*)(ws + off + (size_t)N * FDIM * 4); // N*256 floats

  const int NT = 256;
  const int gemmGrid = (N + 63) / 64;
  const int n4 = N * (FDIM / 4);

  // degrees (init 1.0 for self-loop), then deg^{-1/2}
  fill_kernel<<<(N / 4 + NT - 1) / NT, NT, 0, stream>>>((float4*)dis, 1.0f, N / 4);
  deg_kernel<<<(E + NT - 1) / NT, NT, 0, stream>>>(dst, dis, E);
  rsqrt_kernel<<<(N + NT - 1) / NT, NT, 0, stream>>>(dis, N);

  // h = x @ Wp + bp
  gemm_wmma_f32<<<gemmGrid, NT, 0, stream>>>(x, Wp, bp, h, N);

  for (int layer = 0; layer < 2; ++layer) {
    const float* Wl = (layer == 0) ? W1 : W2;
    const float* bl = (layer == 0) ? b1 : b2;
    fill_kernel<<<(n4 + NT - 1) / NT, NT, 0, stream>>>((float4*)agg, 0.0f, n4);
    gemm_wmma_f32<<<gemmGrid, NT, 0, stream>>>(h, Wl, nullptr, h2, N);
    scatter_kernel<<<8192, NT, 0, stream>>>(src, dst, h2, dis, agg, E);
    epilogue_kernel<<<(n4 + NT - 1) / NT, NT, 0, stream>>>(h, h2, agg, dis, bl, n4,
                                                           layer == 1 ? 1 : 0);
  }
}